// SPMotifNet_68813966016636
// MI455X (gfx1250) — compile-verified
//
#include <hip/hip_runtime.h>

#define N_NODES 100000
#define N_EDGES 1000000
#define HID     64
#define NG      512
#define NC      3
#define NL      3

typedef float v2f __attribute__((ext_vector_type(2)));
typedef float v8f __attribute__((ext_vector_type(8)));
typedef unsigned int u32x4 __attribute__((ext_vector_type(4)));
typedef int i32x4 __attribute__((ext_vector_type(4)));
typedef int i32x8 __attribute__((ext_vector_type(8)));

// ---- Tensor Data Mover: DMA one 64x64 f32 tile (16KB) global -> LDS ----
// D# built per CDNA5 ISA 8.3-8.6: group0 = {count/type/addr}, group1 = dims.
__device__ __forceinline__ void tdm_load_64x64(const float* gsrc, unsigned lds_off) {
  unsigned long long ga = (unsigned long long)(size_t)gsrc;
  u32x4 g0;
  g0[0] = 1u;                                   // count=1, user mode, no gather
  g0[1] = lds_off;                              // lds_addr (bytes)
  g0[2] = (unsigned)ga;                         // global_addr[31:0]
  g0[3] = ((unsigned)(ga >> 32) & 0x01FFFFFFu)  // global_addr[56:32]
          | (2u << 30);                         // type = 2 ("image")
  i32x8 g1;
  g1[0] = 0x00020000;                           // data_size = 2 (4 bytes)
  g1[1] = (int)(64u << 16);                     // tensor_dim0 = 64 (bits 79:48)
  g1[2] = (int)(64u << 16);                     // tensor_dim1 = 64 (bits 111:80)
  g1[3] = (int)(64u << 16);                     // tile_dim0   = 64 (bits 127:112)
  g1[4] = 64;                                   // tile_dim1   = 64 (bits 143:128)
  g1[5] = 64;                                   // tensor_dim0_stride = 64
  g1[6] = 0;
  g1[7] = 0;
  i32x4 z4 = {0, 0, 0, 0};
#if __has_include(<hip/amd_detail/amd_gfx1250_TDM.h>)
  i32x8 z8 = {0, 0, 0, 0, 0, 0, 0, 0};
  __builtin_amdgcn_tensor_load_to_lds(g0, g1, z4, z4, z8, 0);   // 6-arg toolchain
#else
  __builtin_amdgcn_tensor_load_to_lds(g0, g1, z4, z4, 0);       // ROCm 7.2 (5-arg)
#endif
}

// ---------------- utility ----------------
__global__ void k_zero(float* __restrict__ p, int n) {
  int i = blockIdx.x * blockDim.x + threadIdx.x;
  if (i < n) p[i] = 0.0f;
}

// h = x @ We + be   (x:[N,4], We:[4,64])
__global__ void k_embed(const float* __restrict__ x, const float* __restrict__ we,
                        const float* __restrict__ be, float* __restrict__ h) {
  int i = blockIdx.x * blockDim.x + threadIdx.x;
  if (i >= N_NODES * HID) return;
  int n = i >> 6, j = i & 63;
  const float* xr = x + n * 4;
  float acc = be[j];
#pragma unroll
  for (int k = 0; k < 4; ++k) acc += xr[k] * we[k * HID + j];
  h[i] = acc;
}

// ------- fused triple GEMM via fp32 WMMA, weights staged in LDS by TDM -------
// outa = h@w1 + b1 ; outb = h@w2 ; outt = h@w3 + b3  (pre-message accumulator)
// 625 blocks x 10 waves = exactly N_NODES/16 tiles -> EXEC all-1s everywhere.
__global__ __launch_bounds__(320) void k_gemm23(
    const float* __restrict__ h,
    const float* __restrict__ w1, const float* __restrict__ b1,
    const float* __restrict__ w2,
    const float* __restrict__ w3, const float* __restrict__ b3,
    float* __restrict__ outa, float* __restrict__ outb, float* __restrict__ outt) {
  __shared__ float sw[3 * HID * HID];            // 48 KB: W1 | W2 | W3
  if (threadIdx.x < 32) {                        // wave 0 issues the DMAs
    unsigned base = (unsigned)(size_t)(&sw[0]);  // low 32 bits of LDS aperture = offset
    tdm_load_64x64(w1, base);
    tdm_load_64x64(w2, base + 16384u);
    tdm_load_64x64(w3, base + 32768u);
    __builtin_amdgcn_s_wait_tensorcnt(0);
  }
  __syncthreads();

  int wv   = blockIdx.x * 10 + (int)(threadIdx.x >> 5);
  int lane = threadIdx.x & 31;
  int lm   = lane & 15;                 // M (store: N) index within tile
  int kh   = (lane >> 4) << 1;          // 0 (lanes 0-15) or 2 (lanes 16-31)
  int m0   = wv * 16;

  v8f accA[4] = {{}, {}, {}, {}};
  v8f accB[4] = {{}, {}, {}, {}};
  v8f accC[4] = {{}, {}, {}, {}};

  const float* hb = h + (size_t)(m0 + lm) * HID + kh;
  const float* s1 = sw;
  const float* s2 = sw + HID * HID;
  const float* s3 = sw + 2 * HID * HID;

  for (int kk = 0; kk < HID; kk += 4) {
    v2f af = *(const v2f*)(hb + kk);    // A-fragment (16x4 fp32)
    int r0 = (kk + kh) * HID;
    int r1 = r0 + HID;
#pragma unroll
    for (int t = 0; t < 4; ++t) {
      int col = t * 16 + lm;
      v2f f1; f1.x = s1[r0 + col]; f1.y = s1[r1 + col];
      v2f f2; f2.x = s2[r0 + col]; f2.y = s2[r1 + col];
      v2f f3; f3.x = s3[r0 + col]; f3.y = s3[r1 + col];
      accA[t] = __builtin_amdgcn_wmma_f32_16x16x4_f32(false, af, false, f1, (short)0, accA[t], false, false);
      accB[t] = __builtin_amdgcn_wmma_f32_16x16x4_f32(false, af, false, f2, (short)0, accB[t], false, false);
      accC[t] = __builtin_amdgcn_wmma_f32_16x16x4_f32(false, af, false, f3, (short)0, accC[t], false, false);
    }
  }

  int rowoff = (lane >> 4) * 8;          // upper half-wave holds M = r+8
#pragma unroll
  for (int t = 0; t < 4; ++t) {
    int col = t * 16 + lm;
    float ba = b1[col];
    float bc = b3[col];
#pragma unroll
    for (int r = 0; r < 8; ++r) {
      size_t idx = (size_t)(m0 + rowoff + r) * HID + col;
      outa[idx] = accA[t][r] + ba;
      outb[idx] = accB[t][r];
      outt[idx] = accC[t][r] + bc;       // agg pre-seeded with h@W3+b3
    }
  }
}

// ---------------- edge message passing ----------------
// one wave per edge, 2 contiguous floats per lane (coalesced 128B per half-wave)
__global__ void k_edge(const int* __restrict__ src, const int* __restrict__ dst,
                       const float* __restrict__ eattr, const float* __restrict__ eatt,
                       const float* __restrict__ a, const float* __restrict__ b,
                       float* __restrict__ agg) {
  unsigned gid = blockIdx.x * blockDim.x + threadIdx.x;
  int e = (int)(gid >> 5);
  if (e >= N_EDGES) return;
  int lane = (int)(gid & 31);
  float w = eattr[e] * eatt[e];
  size_t si = (size_t)src[e] * HID + lane * 2;
  size_t di = (size_t)dst[e] * HID + lane * 2;
  v2f av = *(const v2f*)(a + si);
  v2f bv = *(const v2f*)(b + di);
  atomicAdd(agg + di,     w * (av.x - bv.x));
  atomicAdd(agg + di + 1, w * (av.y - bv.y));
}

// h = relu(agg) in place (agg already holds msg-sum + h@W3 + b3)
__global__ void k_relu(float* __restrict__ p, int n) {
  int i = blockIdx.x * blockDim.x + threadIdx.x;
  if (i < n) p[i] = fmaxf(p[i], 0.0f);
}

// per-edge attention sums and degrees
__global__ void k_edeg(const int* __restrict__ src, const int* __restrict__ dst,
                       const float* __restrict__ eatt,
                       float* __restrict__ att_sum, float* __restrict__ deg) {
  int e = blockIdx.x * blockDim.x + threadIdx.x;
  if (e >= N_EDGES) return;
  float at = eatt[e];
  int s = src[e], d = dst[e];
  atomicAdd(att_sum + d, at);
  atomicAdd(att_sum + s, at);
  atomicAdd(deg + d, 1.0f);
  atomicAdd(deg + s, 1.0f);
}

// node_att = att_sum / deg (deg==0 -> 1); accumulate per-graph sums
__global__ void k_natt(const float* __restrict__ att_sum, const float* __restrict__ deg,
                       const int* __restrict__ batch,
                       float* __restrict__ node_att, float* __restrict__ g_att,
                       float* __restrict__ g_num) {
  int n = blockIdx.x * blockDim.x + threadIdx.x;
  if (n >= N_NODES) return;
  float dg = deg[n];
  dg = (dg == 0.0f) ? 1.0f : dg;
  float na = att_sum[n] / dg;
  node_att[n] = na;
  int b = batch[n];
  atomicAdd(g_att + b, na);
  atomicAdd(g_num + b, 1.0f);
}

// node_x = h * node_att * (g_num/g_att)[batch]; pooled sum per graph
__global__ void k_pool(const float* __restrict__ h, const float* __restrict__ node_att,
                       const int* __restrict__ batch,
                       const float* __restrict__ g_att, const float* __restrict__ g_num,
                       float* __restrict__ g_sum) {
  unsigned gid = blockIdx.x * blockDim.x + threadIdx.x;
  int n = (int)(gid >> 5);
  if (n >= N_NODES) return;
  int lane = (int)(gid & 31);
  int b = batch[n];
  float ga = g_att[b];
  ga = (ga == 0.0f) ? 1.0f : ga;
  float na = node_att[n] * (g_num[b] / ga);
  size_t idx = (size_t)n * HID + lane * 2;
  v2f hv = *(const v2f*)(h + idx);
  atomicAdd(g_sum + (size_t)b * HID + lane * 2,     na * hv.x);
  atomicAdd(g_sum + (size_t)b * HID + lane * 2 + 1, na * hv.y);
}

// final MLP: relu(graph_x @ fc1 + b1) @ fc2 + b2   (one block per graph)
__global__ void k_mlp(const float* __restrict__ g_sum, const float* __restrict__ g_num,
                      const float* __restrict__ f1w, const float* __restrict__ f1b,
                      const float* __restrict__ f2w, const float* __restrict__ f2b,
                      float* __restrict__ out) {
  __shared__ float gx[HID];
  __shared__ float z1[2 * HID];
  int g = blockIdx.x;
  int t = threadIdx.x;                   // 128 threads
  float cnt = fmaxf(g_num[g], 1.0f);
  if (t < HID) gx[t] = g_sum[(size_t)g * HID + t] / cnt;
  __syncthreads();
  float acc = f1b[t];
  for (int k = 0; k < HID; ++k) acc += gx[k] * f1w[k * (2 * HID) + t];
  z1[t] = fmaxf(acc, 0.0f);
  __syncthreads();
  if (t < NC) {
    float o = f2b[t];
    for (int j = 0; j < 2 * HID; ++j) o += z1[j] * f2w[j * NC + t];
    out[g * NC + t] = o;
  }
}

extern "C" void kernel_launch(void* const* d_in, const int* in_sizes, int n_in,
                              void* d_out, int out_size, void* d_ws, size_t ws_size,
                              hipStream_t stream) {
  (void)in_sizes; (void)n_in; (void)out_size; (void)ws_size;
  const float* x     = (const float*)d_in[0];
  const int*   ei    = (const int*)d_in[1];     // [2,E]: src then dst
  const int*   batch = (const int*)d_in[2];
  const float* eattr = (const float*)d_in[3];
  const float* eatt  = (const float*)d_in[4];
  const float* we    = (const float*)d_in[5];
  const float* be    = (const float*)d_in[6];
  const float* l1w   = (const float*)d_in[7];
  const float* l1b   = (const float*)d_in[8];
  const float* l2w   = (const float*)d_in[9];
  const float* l3w   = (const float*)d_in[10];
  const float* l3b   = (const float*)d_in[11];
  const float* f1w   = (const float*)d_in[12];
  const float* f1b   = (const float*)d_in[13];
  const float* f2w   = (const float*)d_in[14];
  const float* f2b   = (const float*)d_in[15];
  float* out = (float*)d_out;

  const int* src = ei;
  const int* dst = ei + N_EDGES;

  // workspace layout (floats)
  float* ws = (float*)d_ws;
  const size_t NF = (size_t)N_NODES * HID;
  float* B0 = ws;
  float* B1 = B0 + NF;
  float* B2 = B1 + NF;
  float* B3 = B2 + NF;
  float* att_sum  = B3 + NF;
  float* deg      = att_sum + N_NODES;
  float* node_att = deg + N_NODES;
  float* g_att    = node_att + N_NODES;
  float* g_num    = g_att + NG;
  float* g_sum    = g_num + NG;          // NG*HID

  float* ph = B0, *pa = B1, *pb = B2, *pagg = B3;

  k_embed<<<(N_NODES * HID + 255) / 256, 256, 0, stream>>>(x, we, be, ph);

  for (int l = 0; l < NL; ++l) {
    k_gemm23<<<N_NODES / 160, 320, 0, stream>>>(
        ph, l1w + l * HID * HID, l1b + l * HID, l2w + l * HID * HID,
        l3w + l * HID * HID, l3b + l * HID, pa, pb, pagg);
    k_edge<<<(N_EDGES * 32) / 256, 256, 0, stream>>>(src, dst, eattr, eatt, pa, pb, pagg);
    k_relu<<<((int)NF + 255) / 256, 256, 0, stream>>>(pagg, (int)NF);
    float* tmp = ph; ph = pagg; pagg = tmp;   // new h = relu'd accumulator
  }

  // attention + pooling scratch (contiguous region) zero
  int attN = 3 * N_NODES + 2 * NG + NG * HID;
  k_zero<<<(attN + 255) / 256, 256, 0, stream>>>(att_sum, attN);
  k_edeg<<<(N_EDGES + 255) / 256, 256, 0, stream>>>(src, dst, eatt, att_sum, deg);
  k_natt<<<(N_NODES + 255) / 256, 256, 0, stream>>>(att_sum, deg, batch, node_att, g_att, g_num);
  k_pool<<<(N_NODES * 32 + 255) / 256, 256, 0, stream>>>(ph, node_att, batch, g_att, g_num, g_sum);
  k_mlp<<<NG, 128, 0, stream>>>(g_sum, g_num, f1w, f1b, f2w, f2b, out);
}